// MambaBlock_21036749816514
// MI455X (gfx1250) — compile-verified
//
#include <hip/hip_runtime.h>
#include <math.h>

#define D_MODEL 768
#define D_INNER 1536
#define D_STATE 16
#define D_CONV  4
#define DT_RANK 48
#define BATCH   2
#define SEQ     1024
#define NTOK    (BATCH * SEQ)           // 2048 tokens
#define XDBL_W  (DT_RANK + 2 * D_STATE) // 80
#define LN_EPS  1e-5f

typedef __attribute__((ext_vector_type(2))) float v2f;
typedef __attribute__((ext_vector_type(8))) float v8f;

__device__ __forceinline__ float silu_f(float x)    { return x / (1.f + expf(-x)); }
__device__ __forceinline__ float softplus_f(float x){ return (x > 20.f) ? x : log1pf(expf(x)); }

// ---------------------------------------------------------------------------
// C[m,n] = sum_k A[m,k] * W[n,k]   (A: MxK row-major, W: NxK row-major)
// One wave32 computes a 16 x (16*NB) tile via V_WMMA_F32_16X16X4_F32.
// ISA VGPR layouts (CDNA5 ISA §7.12.2):
//   A 16x4 : lanes 0-15 -> M=lane, K = k0+{0,1}; lanes 16-31 -> K = k0+{2,3}
//   B 4x16 : lanes 0-15 -> N=lane, K = k0+{0,1}; lanes 16-31 -> K = k0+{2,3}
//   C/D    : VGPR r, lanes 0-15 -> (M=m0+r, N=lane); lanes 16-31 -> (M=m0+8+r)
// Software-pipelined: fragments for k+4 are loaded before the WMMAs for k are
// issued, so s_wait_loadcnt can tolerate outstanding loads and the matrix
// pipe overlaps with VMEM. global_prefetch_b8 pulls the stream ahead into L2.
// EPI: 0 = plain store, 1 = softplus(acc + bias[n]), 2 = acc + residual[m,n]
// ---------------------------------------------------------------------------
template <int NB, int EPI>
__global__ __launch_bounds__(32)
void wmma_gemm_f32(const float* __restrict__ A, int lda,
                   const float* __restrict__ W, int ldw,
                   float* __restrict__ C, int ldc, int K,
                   const float* __restrict__ bias,
                   const float* __restrict__ residual)
{
    const int lane = threadIdx.x;      // 0..31
    const int hi   = lane >> 4;        // 0: K+{0,1}, 1: K+{2,3}
    const int l15  = lane & 15;
    const int m0   = blockIdx.y << 4;
    const int n0   = blockIdx.x * (16 * NB);

    const float* ap = A + (size_t)(m0 + l15) * lda + 2 * hi;
    const float* wp[NB];
#pragma unroll
    for (int j = 0; j < NB; ++j)
        wp[j] = W + (size_t)(n0 + 16 * j + l15) * ldw + 2 * hi;

    v8f acc[NB] = {};

    // ---- pipeline prologue: stage k = 0 fragments ----
    v2f a_cur = *(const v2f*)(ap);
    v2f b_cur[NB];
#pragma unroll
    for (int j = 0; j < NB; ++j) b_cur[j] = *(const v2f*)(wp[j]);

    // ---- steady state: load k, compute k-4 ----
    for (int k = 4; k < K; k += 4) {
        v2f a_nxt = *(const v2f*)(ap + k);
        v2f b_nxt[NB];
#pragma unroll
        for (int j = 0; j < NB; ++j) b_nxt[j] = *(const v2f*)(wp[j] + k);

        // every 16 k-steps, prefetch one cacheline ahead per stream
        if ((k & 63) == 4) {
            __builtin_prefetch(ap + k + 124, 0, 1);
#pragma unroll
            for (int j = 0; j < NB; ++j) __builtin_prefetch(wp[j] + k + 124, 0, 1);
        }

#pragma unroll
        for (int j = 0; j < NB; ++j)
            acc[j] = __builtin_amdgcn_wmma_f32_16x16x4_f32(
                false, a_cur, false, b_cur[j], (short)0, acc[j], false, false);

        a_cur = a_nxt;
#pragma unroll
        for (int j = 0; j < NB; ++j) b_cur[j] = b_nxt[j];
    }

    // ---- drain last stage ----
#pragma unroll
    for (int j = 0; j < NB; ++j)
        acc[j] = __builtin_amdgcn_wmma_f32_16x16x4_f32(
            false, a_cur, false, b_cur[j], (short)0, acc[j], false, false);

    // ---- epilogue / store ----
    const int crow0 = m0 + hi * 8;
#pragma unroll
    for (int j = 0; j < NB; ++j) {
        const int ccol = n0 + 16 * j + l15;
#pragma unroll
        for (int r = 0; r < 8; ++r) {
            float v = acc[j][r];
            size_t idx = (size_t)(crow0 + r) * ldc + ccol;
            if (EPI == 1)      v = softplus_f(v + bias[ccol]);
            else if (EPI == 2) v += residual[idx];
            C[idx] = v;
        }
    }
}

// ---------------------------------------------------------------------------
// Depthwise causal conv (D_CONV=4) over time + bias + SiLU.
// xb = xz[..., 0:D_INNER] within rows of width 2*D_INNER.
// ---------------------------------------------------------------------------
__global__ __launch_bounds__(256)
void conv_silu_kernel(const float* __restrict__ xz,
                      const float* __restrict__ cw,
                      const float* __restrict__ cb,
                      float* __restrict__ xc)
{
    int idx = blockIdx.x * 256 + threadIdx.x;          // < NTOK * D_INNER
    int d = idx % D_INNER;
    int t = (idx / D_INNER) % SEQ;
    int b = idx / (D_INNER * SEQ);
    float acc = cb[d];
#pragma unroll
    for (int j = 0; j < D_CONV; ++j) {
        int tt = t + j - (D_CONV - 1);
        if (tt >= 0)
            acc = fmaf(xz[((size_t)(b * SEQ + tt)) * (2 * D_INNER) + d],
                       cw[d * D_CONV + j], acc);
    }
    xc[idx] = silu_f(acc);
}

// ---------------------------------------------------------------------------
// Selective scan: one thread per (b, d) channel, h[16] in registers,
// B_t / C_t chunks staged in LDS (shared across the whole block).
// Fuses +u*D and *silu(z) epilogue. 1536 % 256 == 0 -> b uniform per block.
// ---------------------------------------------------------------------------
__global__ __launch_bounds__(256)
void scan_kernel(const float* __restrict__ xdbl,
                 const float* __restrict__ dtb,
                 const float* __restrict__ xc,
                 const float* __restrict__ xz,     // z = xz[..., D_INNER:]
                 const float* __restrict__ A_log,
                 const float* __restrict__ Dvec,
                 float* __restrict__ yact)
{
    const int TCH = 64;
    __shared__ float sB[TCH][D_STATE];
    __shared__ float sC[TCH][D_STATE];

    int gid = blockIdx.x * 256 + threadIdx.x;   // 0 .. BATCH*D_INNER-1
    int b = gid / D_INNER;
    int d = gid % D_INNER;

    float Arow[D_STATE];
#pragma unroll
    for (int s = 0; s < D_STATE; ++s) Arow[s] = -expf(A_log[d * D_STATE + s]);
    float Dd = Dvec[d];
    float h[D_STATE] = {};

    for (int t0 = 0; t0 < SEQ; t0 += TCH) {
        __syncthreads();
        for (int i = threadIdx.x; i < TCH * 2 * D_STATE; i += 256) {
            int tt = i >> 5, c = i & 31;
            float v = xdbl[(size_t)(b * SEQ + t0 + tt) * XDBL_W + DT_RANK + c];
            if (c < D_STATE) sB[tt][c] = v;
            else             sC[tt][c - D_STATE] = v;
        }
        __syncthreads();
        for (int tt = 0; tt < TCH; ++tt) {
            size_t tok  = (size_t)(b * SEQ + t0 + tt);
            size_t base = tok * D_INNER + d;
            float dt = dtb[base];
            float u  = xc[base];
            float du = dt * u;
            float y  = 0.f;
#pragma unroll
            for (int s = 0; s < D_STATE; ++s) {
                h[s] = fmaf(h[s], expf(dt * Arow[s]), du * sB[tt][s]);
                y = fmaf(h[s], sC[tt][s], y);
            }
            y = fmaf(u, Dd, y);
            float z = xz[tok * (2 * D_INNER) + D_INNER + d];
            yact[base] = y * silu_f(z);
        }
    }
}

// ---------------------------------------------------------------------------
// Row LayerNorm over D_MODEL=768, one block per token row.
// ---------------------------------------------------------------------------
__global__ __launch_bounds__(256)
void layernorm_kernel(const float* __restrict__ y,
                      const float* __restrict__ w,
                      const float* __restrict__ bta,
                      float* __restrict__ out)
{
    __shared__ float red[256];
    int row = blockIdx.x;
    const float* yr = y + (size_t)row * D_MODEL;

    float s = 0.f;
    for (int i = threadIdx.x; i < D_MODEL; i += 256) s += yr[i];
    red[threadIdx.x] = s; __syncthreads();
    for (int off = 128; off; off >>= 1) {
        if (threadIdx.x < off) red[threadIdx.x] += red[threadIdx.x + off];
        __syncthreads();
    }
    float mu = red[0] * (1.f / D_MODEL);
    __syncthreads();

    float v = 0.f;
    for (int i = threadIdx.x; i < D_MODEL; i += 256) {
        float dd = yr[i] - mu; v = fmaf(dd, dd, v);
    }
    red[threadIdx.x] = v; __syncthreads();
    for (int off = 128; off; off >>= 1) {
        if (threadIdx.x < off) red[threadIdx.x] += red[threadIdx.x + off];
        __syncthreads();
    }
    float rstd = rsqrtf(red[0] * (1.f / D_MODEL) + LN_EPS);

    for (int i = threadIdx.x; i < D_MODEL; i += 256)
        out[(size_t)row * D_MODEL + i] = (yr[i] - mu) * rstd * w[i] + bta[i];
}

// ---------------------------------------------------------------------------
extern "C" void kernel_launch(void* const* d_in, const int* in_sizes, int n_in,
                              void* d_out, int out_size, void* d_ws, size_t ws_size,
                              hipStream_t stream)
{
    const float* x          = (const float*)d_in[0];
    const float* in_proj_w  = (const float*)d_in[1];
    const float* conv_w     = (const float*)d_in[2];
    const float* conv_b     = (const float*)d_in[3];
    const float* x_proj_w   = (const float*)d_in[4];
    const float* dt_proj_w  = (const float*)d_in[5];
    const float* dt_proj_b  = (const float*)d_in[6];
    const float* A_log      = (const float*)d_in[7];
    const float* Dvec       = (const float*)d_in[8];
    const float* out_proj_w = (const float*)d_in[9];
    const float* ln_w       = (const float*)d_in[10];
    const float* ln_b       = (const float*)d_in[11];
    float* out = (float*)d_out;

    float* ws   = (float*)d_ws;
    float* xz   = ws; ws += (size_t)NTOK * 2 * D_INNER;  // 2048 x 3072
    float* xc   = ws; ws += (size_t)NTOK * D_INNER;      // 2048 x 1536
    float* xdbl = ws; ws += (size_t)NTOK * XDBL_W;       // 2048 x 80
    float* dtb  = ws; ws += (size_t)NTOK * D_INNER;      // 2048 x 1536
    float* yact = ws; ws += (size_t)NTOK * D_INNER;      // 2048 x 1536
    float* ypre = ws; ws += (size_t)NTOK * D_MODEL;      // 2048 x 768

    // 1) in_proj: xz = x @ in_proj_w^T   (2048x3072, K=768)
    wmma_gemm_f32<4, 0><<<dim3((2 * D_INNER) / 64, NTOK / 16), 32, 0, stream>>>(
        x, D_MODEL, in_proj_w, D_MODEL, xz, 2 * D_INNER, D_MODEL, nullptr, nullptr);

    // 2) depthwise causal conv + bias + SiLU -> xc
    conv_silu_kernel<<<(NTOK * D_INNER) / 256, 256, 0, stream>>>(xz, conv_w, conv_b, xc);

    // 3) x_proj: xdbl = xc @ x_proj_w^T  (2048x80, K=1536) — N=80 -> NB=1
    wmma_gemm_f32<1, 0><<<dim3(XDBL_W / 16, NTOK / 16), 32, 0, stream>>>(
        xc, D_INNER, x_proj_w, D_INNER, xdbl, XDBL_W, D_INNER, nullptr, nullptr);

    // 4) dt_proj + bias + softplus: dtb = softplus(xdbl[:, :48] @ dt_proj_w^T + b)
    wmma_gemm_f32<4, 1><<<dim3(D_INNER / 64, NTOK / 16), 32, 0, stream>>>(
        xdbl, XDBL_W, dt_proj_w, DT_RANK, dtb, D_INNER, DT_RANK, dt_proj_b, nullptr);

    // 5) selective scan (+ u*D, * silu(z)) -> yact
    scan_kernel<<<(BATCH * D_INNER) / 256, 256, 0, stream>>>(
        xdbl, dtb, xc, xz, A_log, Dvec, yact);

    // 6) out_proj + residual: ypre = yact @ out_proj_w^T + x  (2048x768, K=1536)
    wmma_gemm_f32<4, 2><<<dim3(D_MODEL / 64, NTOK / 16), 32, 0, stream>>>(
        yact, D_INNER, out_proj_w, D_INNER, ypre, D_MODEL, D_INNER, nullptr, x);

    // 7) LayerNorm -> out
    layernorm_kernel<<<NTOK, 256, 0, stream>>>(ypre, ln_w, ln_b, out);
}